// GraphSAGE_39084202394397
// MI455X (gfx1250) — compile-verified
//
#include <hip/hip_runtime.h>

#define N_NODES 50000
#define N_EDGES 800000
#define DIM 128
#define DIM_OUT 8
#define N_GRAPHS 32

typedef float v2f __attribute__((ext_vector_type(2)));
typedef float v8f __attribute__((ext_vector_type(8)));

// ---------------- degree ----------------
__global__ void deg_kernel(const int* __restrict__ dst, unsigned int* __restrict__ deg) {
    int e = blockIdx.x * blockDim.x + threadIdx.x;
    if (e < N_EDGES) atomicAdd(&deg[dst[e]], 1u);
}

__global__ void invdeg_kernel(const unsigned int* __restrict__ deg, float* __restrict__ invdeg) {
    int n = blockIdx.x * blockDim.x + threadIdx.x;
    if (n < N_NODES) {
        unsigned int d = deg[n];
        invdeg[n] = 1.0f / (float)(d > 0u ? d : 1u);
    }
}

// ---------------- edge scatter (segment_sum of x[src] at dst) ----------------
// one thread per (edge, feature): feature reads coalesced, f32 atomic adds land in L2
__global__ void scatter_kernel(const float* __restrict__ feat, const int* __restrict__ src,
                               const int* __restrict__ dst, float* __restrict__ agg) {
    unsigned int idx = blockIdx.x * blockDim.x + threadIdx.x;
    int e = (int)(idx >> 7);
    int f = (int)(idx & 127u);
    if (e < N_EDGES) {
        int s = src[e];
        int d = dst[e];
        atomicAdd(&agg[(size_t)d * DIM + f], feat[(size_t)s * DIM + f]);
    }
}

// ---------------- SAGE combine: out = relu(agg*invdeg @ Wl + bl + x @ Wr) ----------------
// block = 256 threads (8 waves). Block owns 16 nodes; wave w owns output cols [16w,16w+16).
// Uses V_WMMA_F32_16X16X4_F32 (full fp32) with K-loop of 32 steps over DIM=128, x2 weight mats.
__global__ __launch_bounds__(256) void combine_kernel(
        const float* __restrict__ xin, const float* __restrict__ agg,
        const float* __restrict__ invdeg,
        const float* __restrict__ Wl, const float* __restrict__ bl,
        const float* __restrict__ Wr, float* __restrict__ out) {
    __shared__ float lx[16][132];  // stride 132 -> rows 4 banks apart, conflict-free column reads
    __shared__ float la[16][132];

    const int node0 = blockIdx.x * 16;
    const int tid = threadIdx.x;

    // stage 16x128 tile of x and mean-aggregated neighbors into LDS
    for (int i = 0; i < 8; ++i) {
        int idx = tid + i * 256;
        int r = idx >> 7;
        int f = idx & 127;
        int n = node0 + r;
        float id = invdeg[n];
        lx[r][f] = xin[(size_t)n * DIM + f];
        la[r][f] = agg[(size_t)n * DIM + f] * id;
    }
    __syncthreads();

    const int lane = tid & 31;
    const int w    = tid >> 5;
    const int half = lane >> 4;   // 0: K,K+1  1: K+2,K+3 (A) / rows K,K+1 vs K+2,K+3 (B)
    const int lq   = lane & 15;
    const int c0   = w * 16;

    v8f c = {};
    for (int k = 0; k < DIM; k += 4) {
        int ka = k + 2 * half;
        v2f a, b;
        // term 1: mean-agg @ Wl
        a.x = la[lq][ka];
        a.y = la[lq][ka + 1];
        b.x = Wl[ka * DIM + c0 + lq];
        b.y = Wl[(ka + 1) * DIM + c0 + lq];
        c = __builtin_amdgcn_wmma_f32_16x16x4_f32(false, a, false, b, (short)0, c, false, false);
        // term 2: x @ Wr
        a.x = lx[lq][ka];
        a.y = lx[lq][ka + 1];
        b.x = Wr[ka * DIM + c0 + lq];
        b.y = Wr[(ka + 1) * DIM + c0 + lq];
        c = __builtin_amdgcn_wmma_f32_16x16x4_f32(false, a, false, b, (short)0, c, false, false);
    }

    // D layout: VGPR i -> row M = i + 8*half, col N = lane (lq within half), fuse bias+relu
    const int ncol = c0 + lq;
    const float bias = bl[ncol];
    for (int i = 0; i < 8; ++i) {
        int m = i + 8 * half;
        float v = c[i] + bias;
        out[(size_t)(node0 + m) * DIM + ncol] = v > 0.0f ? v : 0.0f;
    }
}

// ---------------- global mean pool (scatter phase) ----------------
__global__ void pool_kernel(const float* __restrict__ h, const int* __restrict__ batch,
                            float* __restrict__ pooled, float* __restrict__ gcnt) {
    unsigned int idx = blockIdx.x * blockDim.x + threadIdx.x;
    int n = (int)(idx >> 7);
    int f = (int)(idx & 127u);
    if (n < N_NODES) {
        int g = batch[n];
        atomicAdd(&pooled[g * DIM + f], h[(size_t)n * DIM + f]);
        if (f == 0) atomicAdd(&gcnt[g], 1.0f);
    }
}

// ---------------- tiny MLP tail: relu(pooled@W1+b1) @ W2 + b2 ----------------
__global__ __launch_bounds__(256) void mlp_kernel(
        const float* __restrict__ pooled, const float* __restrict__ gcnt,
        const float* __restrict__ W1, const float* __restrict__ b1,
        const float* __restrict__ W2, const float* __restrict__ b2,
        float* __restrict__ out) {
    __shared__ float pm[N_GRAPHS][DIM];
    __shared__ float h1[N_GRAPHS][DIM];
    const int tid = threadIdx.x;

    for (int i = 0; i < 16; ++i) {
        int idx = tid + i * 256;
        int g = idx >> 7;
        int f = idx & 127;
        float cnt = gcnt[g];
        pm[g][f] = pooled[g * DIM + f] / (cnt > 1.0f ? cnt : 1.0f);
    }
    __syncthreads();

    for (int i = 0; i < 16; ++i) {
        int idx = tid + i * 256;
        int g = idx >> 7;
        int f = idx & 127;
        float acc = b1[f];
        for (int k = 0; k < DIM; ++k) acc += pm[g][k] * W1[k * DIM + f];
        h1[g][f] = acc > 0.0f ? acc : 0.0f;
    }
    __syncthreads();

    if (tid < N_GRAPHS * DIM_OUT) {
        int g = tid >> 3;
        int o = tid & 7;
        float acc = b2[o];
        for (int k = 0; k < DIM; ++k) acc += h1[g][k] * W2[k * DIM_OUT + o];
        out[g * DIM_OUT + o] = acc;
    }
}

extern "C" void kernel_launch(void* const* d_in, const int* in_sizes, int n_in,
                              void* d_out, int out_size, void* d_ws, size_t ws_size,
                              hipStream_t stream) {
    const float* x      = (const float*)d_in[0];
    const int*   edge   = (const int*)d_in[1];
    const int*   src    = edge;
    const int*   dst    = edge + N_EDGES;
    const int*   batch  = (const int*)d_in[2];
    const float* Wl1    = (const float*)d_in[3];
    const float* bl1    = (const float*)d_in[4];
    const float* Wr1    = (const float*)d_in[5];
    const float* Wl2    = (const float*)d_in[6];
    const float* bl2    = (const float*)d_in[7];
    const float* Wr2    = (const float*)d_in[8];
    const float* Wl3    = (const float*)d_in[9];
    const float* bl3    = (const float*)d_in[10];
    const float* Wr3    = (const float*)d_in[11];
    const float* W_lin1 = (const float*)d_in[12];
    const float* b_lin1 = (const float*)d_in[13];
    const float* W_lin2 = (const float*)d_in[14];
    const float* b_lin2 = (const float*)d_in[15];
    float* out = (float*)d_out;

    // workspace layout
    float* agg            = (float*)d_ws;              // N*128
    float* hA             = agg + (size_t)N_NODES * DIM;
    float* hB             = hA  + (size_t)N_NODES * DIM;
    float* invdeg         = hB  + (size_t)N_NODES * DIM;
    unsigned int* deg     = (unsigned int*)(invdeg + N_NODES);
    float* pooled         = (float*)(deg + N_NODES);   // 32*128
    float* gcnt           = pooled + N_GRAPHS * DIM;   // 32

    const int TB = 256;
    const int edgeBlocks    = (N_EDGES + TB - 1) / TB;            // 3125
    const int nodeBlocks    = (N_NODES + TB - 1) / TB;            // 196
    const int scatterBlocks = (N_EDGES * DIM) / TB;               // 400000
    const int poolBlocks    = (N_NODES * DIM + TB - 1) / TB;      // 25000
    const int tileBlocks    = N_NODES / 16;                       // 3125 (exact)

    // degrees (shared across layers)
    hipMemsetAsync(deg, 0, N_NODES * sizeof(unsigned int), stream);
    deg_kernel<<<edgeBlocks, TB, 0, stream>>>(dst, deg);
    invdeg_kernel<<<nodeBlocks, TB, 0, stream>>>(deg, invdeg);

    // layer 1: x -> hA
    hipMemsetAsync(agg, 0, (size_t)N_NODES * DIM * sizeof(float), stream);
    scatter_kernel<<<scatterBlocks, TB, 0, stream>>>(x, src, dst, agg);
    combine_kernel<<<tileBlocks, TB, 0, stream>>>(x, agg, invdeg, Wl1, bl1, Wr1, hA);

    // layer 2: hA -> hB
    hipMemsetAsync(agg, 0, (size_t)N_NODES * DIM * sizeof(float), stream);
    scatter_kernel<<<scatterBlocks, TB, 0, stream>>>(hA, src, dst, agg);
    combine_kernel<<<tileBlocks, TB, 0, stream>>>(hA, agg, invdeg, Wl2, bl2, Wr2, hB);

    // layer 3: hB -> hA
    hipMemsetAsync(agg, 0, (size_t)N_NODES * DIM * sizeof(float), stream);
    scatter_kernel<<<scatterBlocks, TB, 0, stream>>>(hB, src, dst, agg);
    combine_kernel<<<tileBlocks, TB, 0, stream>>>(hB, agg, invdeg, Wl3, bl3, Wr3, hA);

    // pool + MLP tail
    hipMemsetAsync(pooled, 0, (N_GRAPHS * DIM + N_GRAPHS) * sizeof(float), stream);
    pool_kernel<<<poolBlocks, TB, 0, stream>>>(hA, batch, pooled, gcnt);
    mlp_kernel<<<1, TB, 0, stream>>>(pooled, gcnt, W_lin1, b_lin1, W_lin2, b_lin2, out);
}